// GCNEncoder_nodeemb_9216999817890
// MI455X (gfx1250) — compile-verified
//
#include <hip/hip_runtime.h>
#include <hip/hip_bf16.h>

typedef float v2f __attribute__((ext_vector_type(2)));
typedef float v8f __attribute__((ext_vector_type(8)));

// ---------------------------------------------------------------------------
// Dense GEMM  C[M,N] = (RELU_IN ? relu(A) : A)[M,K] * W[K,N]   (fp32 WMMA)
// Block: 256 threads = 8 waves; each block covers 128 rows.
// W staged in LDS K-major (transposed) with pitch KP=K+4:
//   - B fragment {W[k][col], W[k+1][col]} = consecutive LDS floats
//     -> one aligned ds_load_b64, immediate offset, zero hot-loop VALU
//   - +4 pad makes the 32-lane b64 access pattern conflict-free
//     (lanes<16 banks {4n+4t,+1}, lanes>=16 banks {4n+4t+2,+3})
// Each wave processes TWO 16-col tiles -> two independent WMMA chains.
// WMMA f32 16x16x4 fragment layouts per CDNA5 ISA 7.12.2.
// ---------------------------------------------------------------------------
template<int K, int N, bool RELU_IN>
__global__ void gemm_wmma(const float* __restrict__ A,
                          const float* __restrict__ W,
                          float* __restrict__ C, int M) {
    static_assert(N % 32 == 0, "paired column tiles");
    constexpr int KP = K + 4;              // padded K pitch (even -> 8B align)
    extern __shared__ float sW[];          // N*KP floats, K-major: sW[j*KP + k]
    const int tid = threadIdx.x;

    // stage W transposed: coalesced global read, scattered LDS write (one-time)
    for (int i = tid; i < K * N; i += 256) {
        const int k = i / N, j = i % N;
        sW[j * KP + k] = W[i];
    }
    __syncthreads();

    const int lane = tid & 31;
    const int wave = tid >> 5;
    const int r0   = blockIdx.x * 128 + wave * 16;
    if (r0 >= M) return;                   // wave-uniform exit (EXEC stays all-1)

    // ---- A fragments for all K, held in registers (K/2 floats per lane) ----
    const int mrow = min(r0 + (lane & 15), M - 1);   // clamp: OOB rows read row M-1
    const int kk   = (lane >> 4) * 2;                // K sub-offset {0,2}
    const float* arow = A + (size_t)mrow * K;
    v2f afrag[K / 4];
#pragma unroll
    for (int t = 0; t < K / 4; ++t) {
        v2f a = *reinterpret_cast<const v2f*>(arow + 4 * t + kk);
        if (RELU_IN) { a.x = fmaxf(a.x, 0.0f); a.y = fmaxf(a.y, 0.0f); }
        afrag[t] = a;
    }

    const int n  = lane & 15;
    const int hi = lane >> 4;

    for (int nt2 = 0; nt2 < N / 32; ++nt2) {
        const int col0 = nt2 * 32 + n;
        const int col1 = col0 + 16;
        const float* bcol0 = sW + col0 * KP + kk;
        const float* bcol1 = sW + col1 * KP + kk;
        v8f acc0 = {};
        v8f acc1 = {};
#pragma unroll
        for (int t = 0; t < K / 4; ++t) {
            const v2f b0 = *reinterpret_cast<const v2f*>(bcol0 + 4 * t);
            const v2f b1 = *reinterpret_cast<const v2f*>(bcol1 + 4 * t);
            acc0 = __builtin_amdgcn_wmma_f32_16x16x4_f32(
                       false, afrag[t], false, b0, (short)0, acc0, false, false);
            acc1 = __builtin_amdgcn_wmma_f32_16x16x4_f32(
                       false, afrag[t], false, b1, (short)0, acc1, false, false);
        }
        const int rowBase = r0 + hi * 8;
#pragma unroll
        for (int r = 0; r < 8; ++r) {
            const int row = rowBase + r;
            if (row < M) {
                C[(size_t)row * N + col0] = acc0[r];
                C[(size_t)row * N + col1] = acc1[r];
            }
        }
    }
}

// ---------------------------------------------------------------------------
// Head GEMM: out = relu(A)[M,K] * [Wm|Wv][K,2*NO] + [bm|bv], split-stored into
// the two tuple outputs (mean, log_var). Same transposed-LDS scheme.
// ---------------------------------------------------------------------------
template<int K, int NO>
__global__ void gemm_head(const float* __restrict__ A,
                          const float* __restrict__ Wm, const float* __restrict__ Wv,
                          const float* __restrict__ bm, const float* __restrict__ bv,
                          float* __restrict__ outm, float* __restrict__ outv, int M) {
    constexpr int N  = 2 * NO;             // 128
    constexpr int KP = K + 4;
    static_assert(N % 32 == 0, "paired column tiles");
    extern __shared__ float smem[];        // N*KP + N floats
    float* sW = smem;                      // K-major: sW[j*KP + k]
    float* sB = smem + N * KP;
    const int tid = threadIdx.x;

    for (int i = tid; i < K * N; i += 256) {
        const int k = i / N, j = i % N;
        sW[j * KP + k] = (j < NO) ? Wm[k * NO + j] : Wv[k * NO + (j - NO)];
    }
    if (tid < N) sB[tid] = (tid < NO) ? bm[tid] : bv[tid - NO];
    __syncthreads();

    const int lane = tid & 31;
    const int wave = tid >> 5;
    const int r0   = blockIdx.x * 128 + wave * 16;
    if (r0 >= M) return;

    const int mrow = min(r0 + (lane & 15), M - 1);
    const int kk   = (lane >> 4) * 2;
    const float* arow = A + (size_t)mrow * K;
    v2f afrag[K / 4];
#pragma unroll
    for (int t = 0; t < K / 4; ++t) {
        v2f a = *reinterpret_cast<const v2f*>(arow + 4 * t + kk);
        a.x = fmaxf(a.x, 0.0f); a.y = fmaxf(a.y, 0.0f);
        afrag[t] = a;
    }

    const int n  = lane & 15;
    const int hi = lane >> 4;

    for (int nt2 = 0; nt2 < N / 32; ++nt2) {
        const int col0 = nt2 * 32 + n;
        const int col1 = col0 + 16;
        const float* bcol0 = sW + col0 * KP + kk;
        const float* bcol1 = sW + col1 * KP + kk;
        v8f acc0 = {};
        v8f acc1 = {};
#pragma unroll
        for (int t = 0; t < K / 4; ++t) {
            const v2f b0 = *reinterpret_cast<const v2f*>(bcol0 + 4 * t);
            const v2f b1 = *reinterpret_cast<const v2f*>(bcol1 + 4 * t);
            acc0 = __builtin_amdgcn_wmma_f32_16x16x4_f32(
                       false, afrag[t], false, b0, (short)0, acc0, false, false);
            acc1 = __builtin_amdgcn_wmma_f32_16x16x4_f32(
                       false, afrag[t], false, b1, (short)0, acc1, false, false);
        }
        const float bias0 = sB[col0];
        const float bias1 = sB[col1];
        const int rowBase = r0 + hi * 8;
#pragma unroll
        for (int r = 0; r < 8; ++r) {
            const int row = rowBase + r;
            if (row < M) {
                const float v0 = acc0[r] + bias0;
                const float v1 = acc1[r] + bias1;
                if (col0 < NO) outm[(size_t)row * NO + col0] = v0;
                else           outv[(size_t)row * NO + (col0 - NO)] = v0;
                if (col1 < NO) outm[(size_t)row * NO + col1] = v1;
                else           outv[(size_t)row * NO + (col1 - NO)] = v1;
            }
        }
    }
}

// ---------------------------------------------------------------------------
// SpMM scatter:  out[b, rows[e], :] += vals[e] * H[b, cols[e], :]
// One wave per edge; float4 gather; f32 atomic scatter (L2-resident tables).
// ---------------------------------------------------------------------------
template<int F>
__global__ void spmm_atomic(const float* __restrict__ H,
                            const int* __restrict__ rows,
                            const int* __restrict__ cols,
                            const float* __restrict__ vals,
                            float* __restrict__ out, int E, int NN) {
    const int lane = threadIdx.x & 31;
    const int wid  = threadIdx.x >> 5;
    const long long e = (long long)blockIdx.x * 8 + wid;
    if (e >= E) return;
    const int   b = blockIdx.y;
    const int   r = rows[e];
    const int   c = cols[e];
    const float v = vals[e];
    const float* hrow = H   + ((size_t)b * NN + c) * F;
    float*       orow = out + ((size_t)b * NN + r) * F;
#pragma unroll
    for (int f0 = lane * 4; f0 < F; f0 += 128) {
        const float4 h4 = *reinterpret_cast<const float4*>(hrow + f0);
        atomicAdd(orow + f0 + 0, v * h4.x);
        atomicAdd(orow + f0 + 1, v * h4.y);
        atomicAdd(orow + f0 + 2, v * h4.z);
        atomicAdd(orow + f0 + 3, v * h4.w);
    }
}

// ---------------------------------------------------------------------------
extern "C" void kernel_launch(void* const* d_in, const int* in_sizes, int n_in,
                              void* d_out, int out_size, void* d_ws, size_t ws_size,
                              hipStream_t stream) {
    (void)n_in; (void)out_size; (void)ws_size;
    constexpr int BATCH = 2, NN = 50000, IN = 128, HID = 256, LAT = 128, OUT = 64;
    constexpr int M = BATCH * NN;
    const int E = in_sizes[1];

    const float* x    = (const float*)d_in[0];
    const int*   rows = (const int*)  d_in[1];
    const int*   cols = (const int*)  d_in[2];
    const float* vals = (const float*)d_in[3];
    const float* W1   = (const float*)d_in[4];
    const float* W2   = (const float*)d_in[5];
    const float* Wm   = (const float*)d_in[6];
    const float* bm   = (const float*)d_in[7];
    const float* Wv   = (const float*)d_in[8];
    const float* bv   = (const float*)d_in[9];

    // workspace layout (ping-pong, 2 x M*HID floats = 204.8 MB)
    float* slot0 = (float*)d_ws;                 // h1 [M,HID]  -> later h2 [M,LAT]
    float* slot1 = slot0 + (size_t)M * HID;      // s1 [M,HID]  -> later s2 [M,LAT]
    float* h1 = slot0;
    float* s1 = slot1;
    float* h2 = slot0;
    float* s2 = slot1;

    float* outm = (float*)d_out;
    float* outv = outm + (size_t)M * OUT;

    const dim3 blk(256);
    const int  gblocks = (M + 127) / 128;
    const dim3 eGrid((E + 7) / 8, BATCH);

    // LDS bytes: transposed W with pitch K+4
    const size_t lds1 = (size_t)HID * (IN  + 4) * sizeof(float);            // 135 KB
    const size_t lds2 = (size_t)LAT * (HID + 4) * sizeof(float);            // 133 KB
    const size_t ldsh = (size_t)(2 * OUT) * (LAT + 4) * sizeof(float)
                      + (size_t)(2 * OUT) * sizeof(float);                  // ~68 KB

    // gc1 linear:  h1 = x @ W1
    gemm_wmma<IN, HID, false><<<gblocks, blk, lds1, stream>>>(x, W1, h1, M);
    // spmm1: s1 = A @ h1   (ReLU deferred into next GEMM's A-read)
    hipMemsetAsync(s1, 0, (size_t)M * HID * sizeof(float), stream);
    spmm_atomic<HID><<<eGrid, blk, 0, stream>>>(h1, rows, cols, vals, s1, E, NN);
    // gc2 linear:  h2 = relu(s1) @ W2
    gemm_wmma<HID, LAT, true><<<gblocks, blk, lds2, stream>>>(s1, W2, h2, M);
    // spmm2: s2 = A @ h2
    hipMemsetAsync(s2, 0, (size_t)M * LAT * sizeof(float), stream);
    spmm_atomic<LAT><<<eGrid, blk, 0, stream>>>(h2, rows, cols, vals, s2, E, NN);
    // heads: mean / log_var = relu(s2) @ [Wm|Wv] + [bm|bv]
    gemm_head<LAT, OUT><<<gblocks, blk, ldsh, stream>>>(
        s2, Wm, Wv, bm, bv, outm, outv, M);
}